// Attention_57818849739014
// MI455X (gfx1250) — compile-verified
//
#include <hip/hip_runtime.h>
#include <hip/hip_bf16.h>

typedef __attribute__((ext_vector_type(2))) float v2f;
typedef __attribute__((ext_vector_type(8))) float v8f;

#define B_  32
#define LK  4096
#define DD  256
#define SS  63

// ---------------------------------------------------------------------------
// Phase 1: e[b,k] = <Q[b], K[b,k]> / 16   via V_WMMA_F32_16X16X4_F32
// One wave computes a 16-key tile, accumulating D/4 = 64 WMMAs.
// A (16x4 f32, 2 VGPRs): lane l<16 -> row l, k={0,1}; lane l+16 -> row l, k={2,3}
// B (4x16 f32, 2 VGPRs): Q[d0+k] broadcast into all 16 columns.
// D (16x16 f32, 8 VGPRs): VGPR j, lanes 0-15 -> M=j; lanes 16-31 -> M=j+8.
// ---------------------------------------------------------------------------
__global__ __launch_bounds__(256) void attn_scores_wmma(
    const float* __restrict__ Q, const float* __restrict__ K,
    float* __restrict__ e) {
  const int lane = threadIdx.x & 31;
  const int wave = threadIdx.x >> 5;
  const int tile = blockIdx.x * 8 + wave;      // B*LK/16 = 8192 tiles
  const int b    = tile >> 8;                  // LK/16 = 256 tiles per batch
  const int k0   = (tile & 255) << 4;
  const int row  = lane & 15;
  const bool hi  = lane >= 16;

  const float4* __restrict__ Krow =
      (const float4*)(K + ((size_t)b * LK + (size_t)(k0 + row)) * DD);
  const float4* __restrict__ Qrow = (const float4*)(Q + (size_t)b * DD);

  v8f c = {};
  #pragma unroll 8
  for (int d4 = 0; d4 < DD / 4; ++d4) {
    float4 kv = Krow[d4];                      // this lane's row, 4 K-columns
    float4 qv = Qrow[d4];                      // broadcast across lanes
    v2f a, bb;
    a.x  = hi ? kv.z : kv.x;
    a.y  = hi ? kv.w : kv.y;
    bb.x = hi ? qv.z : qv.x;
    bb.y = hi ? qv.w : qv.y;
    // 8 args: (neg_a, A, neg_b, B, c_mod, C, reuse_a, reuse_b)
    c = __builtin_amdgcn_wmma_f32_16x16x4_f32(false, a, false, bb,
                                              (short)0, c, false, false);
  }

  const float s = 0.0625f;                     // 1/sqrt(256)
  if (row == 0) {                              // lanes 0 and 16 hold column N=0
    float* ep = e + (size_t)b * LK + k0 + (hi ? 8 : 0);
    ep[0] = c[0] * s; ep[1] = c[1] * s; ep[2] = c[2] * s; ep[3] = c[3] * s;
    ep[4] = c[4] * s; ep[5] = c[5] * s; ep[6] = c[6] * s; ep[7] = c[7] * s;
  }
}

// ---------------------------------------------------------------------------
// Phase 2: per-(batch, segment) softmax over keys (seps[i], seps[i+1]) and
// weighted sum of V rows. One 256-thread block per segment; thread t owns
// output dim d=t, so each V row is read as one fully-coalesced 1KB burst.
// ---------------------------------------------------------------------------
__global__ __launch_bounds__(256) void attn_seg_softmax_v(
    const float* __restrict__ e, const float* __restrict__ V,
    const long long* __restrict__ seps,
    float* __restrict__ y, float* __restrict__ ymask) {
  __shared__ float p[LK];        // exp weights for the segment (max len 4095)
  __shared__ float red[256];

  const int t  = threadIdx.x;
  const int bs = blockIdx.x;                   // b*SS + seg
  const int b  = bs / SS;
  const int sg = bs - b * SS;

  const long long ls = seps[b * (SS + 1) + sg] + 1;   // first key (exclusive sep)
  const long long rs = seps[b * (SS + 1) + sg + 1];   // one past last key
  const int len = (int)(rs - ls);

  float* yrow = y + (size_t)bs * DD;
  if (t == 0) ymask[bs] = 1.0f;
  if (len <= 0) { yrow[t] = 0.0f; return; }    // empty segment -> zero row

  // pass A: stage scores in LDS, block max
  const float* eb = e + (size_t)b * LK + ls;
  float lmax = -INFINITY;
  for (int j = t; j < len; j += 256) {
    float v = eb[j];
    p[j] = v;
    lmax = fmaxf(lmax, v);
  }
  red[t] = lmax; __syncthreads();
  for (int off = 128; off; off >>= 1) {
    if (t < off) red[t] = fmaxf(red[t], red[t + off]);
    __syncthreads();
  }
  const float m = red[0];
  __syncthreads();

  // pass B: exp and block sum
  float lsum = 0.f;
  for (int j = t; j < len; j += 256) {
    float ex = __expf(p[j] - m);
    p[j] = ex;
    lsum += ex;
  }
  red[t] = lsum; __syncthreads();
  for (int off = 128; off; off >>= 1) {
    if (t < off) red[t] += red[t + off];
    __syncthreads();
  }
  const float den = fmaxf(red[0], 1e-30f);

  // pass C: y[d=t] = sum_j p[j] * V[b, ls+j, t] / den  (coalesced 1KB rows)
  const float* vp = V + ((size_t)b * LK + (size_t)ls) * DD + t;
  float acc = 0.f;
  int j = 0;
  for (; j + 4 <= len; j += 4) {
    acc = fmaf(p[j    ], vp[(size_t)(j    ) * DD], acc);
    acc = fmaf(p[j + 1], vp[(size_t)(j + 1) * DD], acc);
    acc = fmaf(p[j + 2], vp[(size_t)(j + 2) * DD], acc);
    acc = fmaf(p[j + 3], vp[(size_t)(j + 3) * DD], acc);
  }
  for (; j < len; ++j) acc = fmaf(p[j], vp[(size_t)j * DD], acc);
  yrow[t] = acc / den;
}

extern "C" void kernel_launch(void* const* d_in, const int* in_sizes, int n_in,
                              void* d_out, int out_size, void* d_ws, size_t ws_size,
                              hipStream_t stream) {
  const float*     Q    = (const float*)d_in[0];      // [B,1,D] f32
  const float*     K    = (const float*)d_in[1];      // [B,Lk,D] f32
  const float*     V    = (const float*)d_in[2];      // [B,Lk,D] f32
  const long long* seps = (const long long*)d_in[3];  // [B,S+1] i64

  float* e     = (float*)d_ws;                        // B*LK floats = 512 KB
  float* y     = (float*)d_out;                       // [B,S,D]
  float* ymask = y + (size_t)B_ * SS * DD;            // [B,S]

  // Phase 1: 8192 key-tiles, 8 waves/block -> 1024 blocks
  attn_scores_wmma<<<(B_ * LK) / (16 * 8), 256, 0, stream>>>(Q, K, e);
  // Phase 2: one block per (batch, segment)
  attn_seg_softmax_v<<<B_ * SS, 256, 0, stream>>>(e, V, seps, y, ymask);
}